// ResponseSimpleBaselineProt_54142357733865
// MI455X (gfx1250) — compile-verified
//
#include <hip/hip_runtime.h>

typedef __attribute__((ext_vector_type(2))) float v2f;
typedef __attribute__((ext_vector_type(8))) float v8f;

#define B_SZ 4096
#define P_SZ 19000
#define T_SZ 32
#define D_SZ 4000
#define H1   2048
#define H2   1024

// Special W1 rows: [prot1 (0..P-1)][conc_slotA (P)][prot2 (P+1..2P)][conc_slotB (2P+1)]
#define ROW_CA ((size_t)P_SZ)           // 19000
#define ROW_P2 ((size_t)(P_SZ + 1))     // 19001 : base of second protein block
#define ROW_CB ((size_t)(2 * P_SZ + 1)) // 38001

// ---------------------------------------------------------------------------
// Kernel 1: per-drug embeddings.  E1[d] = sum_{unique p in tgt(d)} W1[p,:]
//                                 E2[d] = sum_{unique p in tgt(d)} W1[P+1+p,:]
// One block per drug; 256 threads x 8 cols (2 float4s) = 2048 cols.
// ---------------------------------------------------------------------------
__global__ __launch_bounds__(256) void embed_kernel(
    const int* __restrict__ tgts, const float* __restrict__ W1,
    float* __restrict__ E1, float* __restrict__ E2)
{
    __shared__ int s_tgt[T_SZ];
    __shared__ int s_keep[T_SZ];
    const int d = blockIdx.x;

    if (threadIdx.x < T_SZ) s_tgt[threadIdx.x] = tgts[d * T_SZ + threadIdx.x];
    __syncthreads();
    if (threadIdx.x < T_SZ) {
        const int t = s_tgt[threadIdx.x];
        int keep = 1;
        for (int u = 0; u < (int)threadIdx.x; ++u) keep &= (s_tgt[u] != t);
        s_keep[threadIdx.x] = keep;  // duplicates in the scatter only set 1 once
    }
    __syncthreads();

    const int c4 = threadIdx.x;  // float4 index: cols c4*4 and 1024 + c4*4
    float4 a0 = make_float4(0.f, 0.f, 0.f, 0.f), a1 = a0, e0 = a0, e1 = a0;

    for (int t = 0; t < T_SZ; ++t) {
        if (!s_keep[t]) continue;
        const size_t r = (size_t)s_tgt[t];
        const float4* r1 = (const float4*)(W1 + r * H1);
        const float4* r2 = (const float4*)(W1 + (ROW_P2 + r) * H1);
        float4 x;
        x = r1[c4];        a0.x += x.x; a0.y += x.y; a0.z += x.z; a0.w += x.w;
        x = r1[c4 + 256];  a1.x += x.x; a1.y += x.y; a1.z += x.z; a1.w += x.w;
        x = r2[c4];        e0.x += x.x; e0.y += x.y; e0.z += x.z; e0.w += x.w;
        x = r2[c4 + 256];  e1.x += x.x; e1.y += x.y; e1.z += x.z; e1.w += x.w;
    }
    float4* o1 = (float4*)(E1 + (size_t)d * H1);
    float4* o2 = (float4*)(E2 + (size_t)d * H1);
    o1[c4] = a0; o1[c4 + 256] = a1;
    o2[c4] = e0; o2[c4 + 256] = e1;
}

// ---------------------------------------------------------------------------
// Kernel 2: fused layer1 (sparse assembly) + layer2 (f32 WMMA) + layer3 (dot).
// Block = 16 output rows (rows 0..4095 = order_1, 4096..8191 = order_2).
// 8 waves; wave w owns N-tiles [w*128, w*128+128) => 8 x v8f accumulators.
// K processed in 4 chunks of 512 staged through LDS (padded stride 516).
// ---------------------------------------------------------------------------
#define KCH  512
#define LDST (KCH + 4)  // pad => conflict-free ds_load_b64 across 16 rows
#define NT   8

__global__ __launch_bounds__(256) void fused_kernel(
    const int* __restrict__ pairs, const float* __restrict__ conc,
    const float* __restrict__ W1, const float* __restrict__ b1,
    const float* __restrict__ W2, const float* __restrict__ b2,
    const float* __restrict__ W3, const float* __restrict__ b3,
    const float* __restrict__ E1, const float* __restrict__ E2,
    float* __restrict__ out)
{
    __shared__ float sH[16 * LDST];   // 33 KB hidden tile (fp32, chunked K)
    __shared__ float sRed[256 * 8];   // 8 KB layer-3 partials
    __shared__ int   s_dA[16], s_dB[16];
    __shared__ float s_cA[16], s_cB[16];

    const int m0   = blockIdx.x * 16;
    const int lane = threadIdx.x & 31;
    const int wave = threadIdx.x >> 5;
    const int half = lane >> 4;    // 0: K rows {0,1}, 1: K rows {2,3}
    const int mrow = lane & 15;

    if (threadIdx.x < 16) {
        const int m   = m0 + threadIdx.x;
        const int b   = m & (B_SZ - 1);
        const int ord = m >> 12;                    // 0 = order_1, 1 = order_2
        const int   d1 = pairs[2 * b], d2 = pairs[2 * b + 1];
        const float c1 = conc[2 * b],  c2 = conc[2 * b + 1];
        s_dA[threadIdx.x] = ord ? d2 : d1;
        s_dB[threadIdx.x] = ord ? d1 : d2;
        s_cA[threadIdx.x] = ord ? c2 : c1;
        s_cB[threadIdx.x] = ord ? c1 : c2;
    }

    v8f acc[NT] = {};

    for (int ch = 0; ch < 4; ++ch) {
        const int kbase = ch * KCH;
        __syncthreads();  // meta ready (ch 0) / previous chunk's reads done

        // ---- layer 1: fill 16 x 512 hidden tile (relu(b1 + E1 + E2 + conc rows))
        #pragma unroll
        for (int it = 0; it < 8; ++it) {
            const int e   = it * 256 + threadIdx.x;  // 0..2047 float4 slots
            const int r   = e >> 7;                  // row 0..15
            const int cc  = (e & 127) << 2;          // col in chunk (mult of 4)
            const int col = kbase + cc;
            const float4 bb = *(const float4*)(b1 + col);
            const float4 ea = *(const float4*)(E1 + (size_t)s_dA[r] * H1 + col);
            const float4 eb = *(const float4*)(E2 + (size_t)s_dB[r] * H1 + col);
            const float4 wa = *(const float4*)(W1 + ROW_CA * H1 + col);
            const float4 wb = *(const float4*)(W1 + ROW_CB * H1 + col);
            const float cA = s_cA[r], cB = s_cB[r];
            float4 h;
            h.x = fmaxf(bb.x + ea.x + eb.x + cA * wa.x + cB * wb.x, 0.f);
            h.y = fmaxf(bb.y + ea.y + eb.y + cA * wa.y + cB * wb.y, 0.f);
            h.z = fmaxf(bb.z + ea.z + eb.z + cA * wa.z + cB * wb.z, 0.f);
            h.w = fmaxf(bb.w + ea.w + eb.w + cA * wa.w + cB * wb.w, 0.f);
            *(float4*)(sH + r * LDST + cc) = h;
        }
        __syncthreads();

        // ---- layer 2: fp32 WMMA K-loop over this chunk
        for (int kk = 0; kk < KCH; kk += 4) {
            // A 16x4: lanes 0-15 hold K={kk,kk+1}, lanes 16-31 K={kk+2,kk+3}
            const v2f A = *(const v2f*)(sH + mrow * LDST + kk + half * 2);
            const int kg = kbase + kk + half * 2;
            const float* w2p = W2 + (size_t)kg * H2 + wave * (NT * 16) + mrow;
            #pragma unroll
            for (int t = 0; t < NT; ++t) {
                v2f Bv;                       // B 4x16: row kg (vgpr0), kg+1 (vgpr1)
                Bv.x = w2p[t * 16];
                Bv.y = w2p[H2 + t * 16];
                acc[t] = __builtin_amdgcn_wmma_f32_16x16x4_f32(
                    false, A, false, Bv, (short)0, acc[t], false, false);
            }
        }
    }

    // ---- layer 3: +b2, relu, dot with W3 (per-lane partials, fixed-order reduce)
    float p[8];
    #pragma unroll
    for (int j = 0; j < 8; ++j) p[j] = 0.f;
    #pragma unroll
    for (int t = 0; t < NT; ++t) {
        const int n   = wave * (NT * 16) + t * 16 + mrow;
        const float w3  = W3[n];
        const float bb2 = b2[n];
        #pragma unroll
        for (int j = 0; j < 8; ++j) {   // C/D: vgpr j = row j + 8*half, col = mrow
            const float h2 = fmaxf(acc[t][j] + bb2, 0.f);
            p[j] = fmaf(h2, w3, p[j]);
        }
    }
    #pragma unroll
    for (int j = 0; j < 8; ++j) sRed[threadIdx.x * 8 + j] = p[j];
    __syncthreads();

    if (threadIdx.x < 16) {
        const int r = threadIdx.x;          // output row m0 + r
        const int j = r & 7, hf = r >> 3;   // r = j + 8*hf
        float s = b3[0];
        for (int w = 0; w < 8; ++w)
            for (int l = 0; l < 16; ++l)
                s += sRed[(w * 32 + hf * 16 + l) * 8 + j];
        out[m0 + r] = s;                    // out = [order_1 (4096) | order_2 (4096)]
    }
}

// ---------------------------------------------------------------------------
extern "C" void kernel_launch(void* const* d_in, const int* in_sizes, int n_in,
                              void* d_out, int out_size, void* d_ws, size_t ws_size,
                              hipStream_t stream) {
    const int*   pairs = (const int*)d_in[0];    // [B,2]
    const int*   tgts  = (const int*)d_in[1];    // [D,T]
    const float* conc  = (const float*)d_in[2];  // [B,2]
    const float* W1    = (const float*)d_in[3];  // [38002,2048]
    const float* b1    = (const float*)d_in[4];  // [2048]
    const float* W2    = (const float*)d_in[5];  // [2048,1024]
    const float* b2    = (const float*)d_in[6];  // [1024]
    const float* W3    = (const float*)d_in[7];  // [1024,1]
    const float* b3    = (const float*)d_in[8];  // [1]
    float* out = (float*)d_out;                  // [2*B]

    float* E1 = (float*)d_ws;                    // [D, H1]
    float* E2 = E1 + (size_t)D_SZ * H1;          // [D, H1]  (total 65.5 MB)

    embed_kernel<<<D_SZ, 256, 0, stream>>>(tgts, W1, E1, E2);
    fused_kernel<<<(2 * B_SZ) / 16, 256, 0, stream>>>(
        pairs, conc, W1, b1, W2, b2, W3, b3, E1, E2, out);
}